// DeformableFusionLayer_27857157882352
// MI455X (gfx1250) — compile-verified
//
#include <hip/hip_runtime.h>
#include <hip/hip_bf16.h>

typedef float v2f __attribute__((ext_vector_type(2)));
typedef float v8f __attribute__((ext_vector_type(8)));

#define BB    8
#define CIN   256
#define COUT  256
#define HH    64
#define WW    64
#define HW    (HH * WW)          // 4096
#define NTOT  (BB * HW)          // 32768
#define KK9   9
#define EPSV  1e-5f

// ---------------------------------------------------------------------------
// Kernel 1: offset conv  (B,256,64,64) -> off[b][j][p], j = kk*2 + {dy,dx}
// ---------------------------------------------------------------------------
__global__ void deform_offset_conv(const float* __restrict__ x,
                                   const float* __restrict__ w_off,
                                   const float* __restrict__ b_off,
                                   float* __restrict__ off) {
  int gid = blockIdx.x * blockDim.x + threadIdx.x;   // 0 .. 32767
  int b  = gid >> 12;
  int p  = gid & 4095;
  int h  = p >> 6;
  int w_ = p & 63;

  float acc[18];
#pragma unroll
  for (int j = 0; j < 18; ++j) acc[j] = b_off[j];

  const float* xb = x + (size_t)b * CIN * HW;
  for (int c = 0; c < CIN; ++c) {
    float patch[9];
    const float* xc = xb + (size_t)c * HW;
#pragma unroll
    for (int dy = -1; dy <= 1; ++dy) {
#pragma unroll
      for (int dx = -1; dx <= 1; ++dx) {
        int yy = h + dy, xx = w_ + dx;
        bool ok = (yy >= 0) & (yy < HH) & (xx >= 0) & (xx < WW);
        patch[(dy + 1) * 3 + (dx + 1)] = ok ? xc[yy * WW + xx] : 0.0f;
      }
    }
#pragma unroll
    for (int j = 0; j < 18; ++j) {
      const float* wp = w_off + ((size_t)j * CIN + c) * 9;   // uniform -> SMEM
#pragma unroll
      for (int q = 0; q < 9; ++q) acc[j] = fmaf(wp[q], patch[q], acc[j]);
    }
  }
#pragma unroll
  for (int j = 0; j < 18; ++j)
    off[((size_t)b * 18 + j) * HW + p] = acc[j];
}

// ---------------------------------------------------------------------------
// Kernel 2: repack weights (O,C,3,3) -> wt[kk][c][o]  (o contiguous)
// ---------------------------------------------------------------------------
__global__ void deform_repack_w(const float* __restrict__ w,
                                float* __restrict__ wt) {
  int gid = blockIdx.x * blockDim.x + threadIdx.x;   // 0 .. 589823
  int o  = gid & 255;
  int c  = (gid >> 8) & 255;
  int kk = gid >> 16;
  wt[gid] = w[((size_t)o * CIN + c) * 9 + kk];
}

// ---------------------------------------------------------------------------
// Kernel 3: fused bilinear gather + GEMM via V_WMMA_F32_16X16X4_F32
//   out_pre[o][n] = b[o] + sum_{kk,c} wt[kk][c][o] * sampled[c][kk][n]
//   block: 256 threads (8 waves), tile M=256 (all Cout) x N=64 (one row).
//   A tile [o][34] and B tile [n][34]: (k,k+1) adjacent + even index
//   -> every WMMA operand is a single aligned ds_load_b64.
// ---------------------------------------------------------------------------
__global__ void __launch_bounds__(256)
deform_gemm_wmma(const float* __restrict__ x,
                 const float* __restrict__ off,
                 const float* __restrict__ wt,
                 const float* __restrict__ bias,
                 float* __restrict__ pre) {
  __shared__ float As[256 * 34];      // 34816 B  weights [o][c-chunk]
  __shared__ float Bs[64 * 34];       //  8704 B  samples [n][c-chunk]
  __shared__ int   sIdx[4 * 64];      // 4 corner flat indices per n
  __shared__ float sWt[4 * 64];       // 4 validity-weighted bilinear weights

  const int tid   = threadIdx.x;
  const int lane  = tid & 31;
  const int wave  = tid >> 5;
  const int n_base = blockIdx.x * 64;          // global spatial index base
  const int b_img  = n_base >> 12;             // whole tile in one image
  const int p_base = n_base & 4095;            // one full image row (W=64)

  const float* xb = x + (size_t)b_img * CIN * HW;

  const int row16 = lane & 15;
  const int ksel  = (lane >> 4) << 1;          // 0 for lanes 0-15, 2 for 16-31
  const int mBase = wave << 5;                 // 32 Cout rows per wave

  v8f acc[8];
#pragma unroll
  for (int t = 0; t < 8; ++t) acc[t] = (v8f){0.f,0.f,0.f,0.f,0.f,0.f,0.f,0.f};

  for (int kk = 0; kk < KK9; ++kk) {
    // ---- per-tap sampling coefficients for the 64 positions ----
    __syncthreads();
    if (tid < 64) {
      const int n  = tid;
      const int p  = p_base + n;
      const int h  = p >> 6;
      const int w_ = p & 63;
      const float dy = off[((size_t)b_img * 18 + 2 * kk) * HW + p];
      const float dx = off[((size_t)b_img * 18 + 2 * kk + 1) * HW + p];
      const float py = (float)(h + kk / 3 - 1) + dy;
      const float px = (float)(w_ + kk % 3 - 1) + dx;
      const float y0f = floorf(py), x0f = floorf(px);
      const float fy = py - y0f, fx = px - x0f;
      const int y0 = (int)y0f, x0 = (int)x0f;
#pragma unroll
      for (int cr = 0; cr < 4; ++cr) {
        const int yy = y0 + (cr >> 1);
        const int xx = x0 + (cr & 1);
        const float wy = (cr >> 1) ? fy : (1.0f - fy);
        const float wx = (cr & 1) ? fx : (1.0f - fx);
        const bool ok = (yy >= 0) & (yy <= HH - 1) & (xx >= 0) & (xx <= WW - 1);
        const int yc = min(max(yy, 0), HH - 1);
        const int xc = min(max(xx, 0), WW - 1);
        sIdx[cr * 64 + n] = yc * WW + xc;
        sWt[cr * 64 + n]  = ok ? (wy * wx) : 0.0f;
      }
    }
    __syncthreads();

    for (int c0 = 0; c0 < CIN; c0 += 32) {
      // ---- stage A: weights wt[kk][c0..c0+31][o], coalesced over o ----
      {
        const float* wp = wt + ((size_t)kk * CIN + c0) * COUT + tid;
        __builtin_prefetch(wp + 32 * COUT, 0, 1);   // next chunk (speculative)
#pragma unroll 8
        for (int j = 0; j < 32; ++j) As[tid * 34 + j] = wp[j * COUT];
      }
      // ---- stage B: bilinear-gathered samples, lane<->n for coalescing ----
      {
        const int n    = tid & 63;
        const int csub = tid >> 6;                       // 0..3
        const int i0 = sIdx[n],       i1 = sIdx[64 + n];
        const int i2 = sIdx[128 + n], i3 = sIdx[192 + n];
        const float g0 = sWt[n],       g1 = sWt[64 + n];
        const float g2 = sWt[128 + n], g3 = sWt[192 + n];
#pragma unroll
        for (int i = 0; i < 8; ++i) {
          const int c = c0 + csub + i * 4;
          const float* xc = xb + (size_t)c * HW;
          const float g = g0 * xc[i0] + g1 * xc[i1] + g2 * xc[i2] + g3 * xc[i3];
          Bs[n * 34 + csub + i * 4] = g;
        }
      }
      __syncthreads();

      // ---- WMMA over the 32-deep K chunk: 8 x (16x16x4) tiles per wave ----
#pragma unroll
      for (int k4 = 0; k4 < 32; k4 += 4) {
        const int ka = k4 + ksel;                        // always even
        const v2f a0 = *(const v2f*)&As[(mBase + row16) * 34 + ka];
        const v2f a1 = *(const v2f*)&As[(mBase + 16 + row16) * 34 + ka];
#pragma unroll
        for (int nj = 0; nj < 4; ++nj) {
          const v2f bb = *(const v2f*)&Bs[(nj * 16 + row16) * 34 + ka];
          acc[nj] = __builtin_amdgcn_wmma_f32_16x16x4_f32(
              false, a0, false, bb, (short)0, acc[nj], false, false);
          acc[4 + nj] = __builtin_amdgcn_wmma_f32_16x16x4_f32(
              false, a1, false, bb, (short)0, acc[4 + nj], false, false);
        }
      }
      __syncthreads();
    }
  }

  // ---- epilogue: D layout (VGPR r: lanes0-15 M=r, lanes16-31 M=r+8) ----
#pragma unroll
  for (int t = 0; t < 8; ++t) {
    const int mi = (t >> 2) << 4;                // 0 or 16
    const int nj = t & 3;
    const int oB = mBase + mi + ((lane >> 4) << 3);
    const int n  = n_base + nj * 16 + row16;
#pragma unroll
    for (int r = 0; r < 8; ++r) {
      const int o = oB + r;
      pre[(size_t)o * NTOT + n] = acc[t][r] + bias[o];
    }
  }
}

// ---------------------------------------------------------------------------
// Kernel 4: per-channel mean / rstd over 32768 elements (float4 reads)
// ---------------------------------------------------------------------------
__global__ void deform_stats(const float* __restrict__ pre,
                             float* __restrict__ stats) {
  __shared__ float r1[256];
  __shared__ float r2[256];
  const int o = blockIdx.x;
  const int tid = threadIdx.x;
  const float4* po = (const float4*)(pre + (size_t)o * NTOT);
  float s = 0.f, s2 = 0.f;
  for (int i = tid; i < NTOT / 4; i += 256) {
    const float4 v = po[i];
    s  += v.x + v.y + v.z + v.w;
    s2 += v.x * v.x + v.y * v.y + v.z * v.z + v.w * v.w;
  }
  r1[tid] = s; r2[tid] = s2;
  __syncthreads();
  for (int stride = 128; stride > 0; stride >>= 1) {
    if (tid < stride) { r1[tid] += r1[tid + stride]; r2[tid] += r2[tid + stride]; }
    __syncthreads();
  }
  if (tid == 0) {
    const float mean = r1[0] * (1.0f / NTOT);
    const float var  = r2[0] * (1.0f / NTOT) - mean * mean;
    stats[o]        = mean;
    stats[256 + o]  = rsqrtf(var + EPSV);
  }
}

// ---------------------------------------------------------------------------
// Kernel 5: batchnorm affine + ReLU -> NCHW output (float4 in/out)
// ---------------------------------------------------------------------------
__global__ void deform_bn_relu(const float* __restrict__ pre,
                               const float* __restrict__ stats,
                               const float* __restrict__ gamma,
                               const float* __restrict__ beta,
                               float* __restrict__ out) {
  const int gid = blockIdx.x * blockDim.x + threadIdx.x;  // 0 .. 2097151 (x4)
  const int p4   = gid & 1023;                 // float4 index within image row-plane
  const int o    = (gid >> 10) & 255;
  const int bimg = gid >> 18;
  const float4 v = *(const float4*)(pre + (size_t)o * NTOT + bimg * HW + p4 * 4);
  const float mu = stats[o];
  const float rs = stats[256 + o];
  const float g  = gamma[o];
  const float be = beta[o];
  float4 y;
  y.x = fmaxf(g * (v.x - mu) * rs + be, 0.0f);
  y.y = fmaxf(g * (v.y - mu) * rs + be, 0.0f);
  y.z = fmaxf(g * (v.z - mu) * rs + be, 0.0f);
  y.w = fmaxf(g * (v.w - mu) * rs + be, 0.0f);
  *(float4*)(out + (size_t)gid * 4) = y;
}

// ---------------------------------------------------------------------------
extern "C" void kernel_launch(void* const* d_in, const int* in_sizes, int n_in,
                              void* d_out, int out_size, void* d_ws, size_t ws_size,
                              hipStream_t stream) {
  const float* x     = (const float*)d_in[0];
  const float* w_off = (const float*)d_in[1];
  const float* b_off = (const float*)d_in[2];
  const float* w     = (const float*)d_in[3];
  const float* bias  = (const float*)d_in[4];
  const float* gamma = (const float*)d_in[5];
  const float* beta  = (const float*)d_in[6];
  float* out = (float*)d_out;

  float* ws    = (float*)d_ws;
  float* off   = ws;                          //   589,824 floats
  float* wt    = ws + 589824;                 //   589,824 floats
  float* pre   = ws + 1179648;                // 8,388,608 floats
  float* stats = ws + 9568256;                //       512 floats

  deform_offset_conv<<<NTOT / 256, 256, 0, stream>>>(x, w_off, b_off, off);
  deform_repack_w<<<(KK9 * CIN * COUT) / 256, 256, 0, stream>>>(w, wt);
  deform_gemm_wmma<<<NTOT / 64, 256, 0, stream>>>(x, off, wt, bias, pre);
  deform_stats<<<COUT, 256, 0, stream>>>(pre, stats);
  deform_bn_relu<<<(BB * COUT * HW) / 1024, 256, 0, stream>>>(pre, stats, gamma, beta, out);
}